// Net_33775622816473
// MI455X (gfx1250) — compile-verified
//
#include <hip/hip_runtime.h>
#include <hip/hip_bf16.h>

typedef __attribute__((ext_vector_type(2))) float v2f;
typedef __attribute__((ext_vector_type(8))) float v8f;

#define IN_DIM 512
#define FDIM   16

// ---------------------------------------------------------------------------
// 1) init: deg starts at 1.0 (self-loop), zero both aggregation accumulators.
//    Must run every call: harness replays the graph and atomics accumulate.
// ---------------------------------------------------------------------------
__global__ void init_ws_kernel(float* __restrict__ dinv,
                               float* __restrict__ a1,
                               float* __restrict__ a2, int N) {
    int i = blockIdx.x * blockDim.x + threadIdx.x;
    if (i < N) dinv[i] = 1.0f;          // self-loop contributes 1 to in-degree
    if (i < N * FDIM) { a1[i] = 0.0f; a2[i] = 0.0f; }
}

// ---------------------------------------------------------------------------
// 2) in-degree over destination (col) nodes
// ---------------------------------------------------------------------------
__global__ void degree_kernel(const long long* __restrict__ ei, long long E,
                              float* __restrict__ deg) {
    long long e = (long long)blockIdx.x * blockDim.x + threadIdx.x;
    if (e < E) atomicAdd(&deg[(int)ei[E + e]], 1.0f);
}

// ---------------------------------------------------------------------------
// 3) deg -> deg^{-1/2} (deg >= 1 always, self-loop included)
// ---------------------------------------------------------------------------
__global__ void rsqrt_kernel(float* __restrict__ dinv, int N) {
    int i = blockIdx.x * blockDim.x + threadIdx.x;
    if (i < N) dinv[i] = rsqrtf(dinv[i]);
}

// ---------------------------------------------------------------------------
// 4) h = x @ W1  via V_WMMA_F32_16X16X4_F32 (fp32 matrix pipe, exact fp32).
//    8 waves/block; each wave owns one 16-row tile; 128 WMMA steps over K=512.
//    A layout (16x4 f32): lanes 0-15 -> K = k,k+1 ; lanes 16-31 -> K = k+2,k+3
//    (one 8-byte global load per lane per step).  B staged in LDS (32 KB).
//    C/D layout: VGPR r -> row r (lanes 0-15) / row r+8 (lanes 16-31).
// ---------------------------------------------------------------------------
__global__ __launch_bounds__(256)
void gemm1_kernel(const float* __restrict__ x, const float* __restrict__ W,
                  float* __restrict__ h, int N) {
    __shared__ float sW[IN_DIM * FDIM];
    for (int i = threadIdx.x; i < IN_DIM * FDIM; i += 256) sW[i] = W[i];
    __syncthreads();

    const int wave = threadIdx.x >> 5;
    const int lane = threadIdx.x & 31;
    const int ntiles = (N + 15) >> 4;
    const int tile = blockIdx.x * 8 + wave;
    if (tile >= ntiles) return;                 // uniform per wave: EXEC stays all-1s

    const int mn   = lane & 15;                 // A: row within tile, B: output col
    const int koff = (lane >> 4) << 1;          // 0 or 2
    int row = tile * 16 + mn; if (row > N - 1) row = N - 1;
    const float* xp = x + (size_t)row * IN_DIM + koff;

    v8f c = {};
    for (int k = 0; k < IN_DIM; k += 4) {
        v2f a = *(const v2f*)(xp + k);                    // K = k+koff, k+koff+1
        v2f b;
        b.x = sW[(k + koff)     * FDIM + mn];             // W1[K][n]
        b.y = sW[(k + koff + 1) * FDIM + mn];
        c = __builtin_amdgcn_wmma_f32_16x16x4_f32(
                /*neg_a=*/false, a, /*neg_b=*/false, b,
                /*c_mod=*/(short)0, c, /*reuse_a=*/false, /*reuse_b=*/false);
    }

    const int m0 = (lane < 16) ? 0 : 8;
    for (int r = 0; r < 8; ++r) {
        int gm = tile * 16 + m0 + r;
        if (gm < N) h[(size_t)gm * FDIM + mn] = c[r];
    }
}

// ---------------------------------------------------------------------------
// 7) h2 = z @ W2  (K = 16) — same WMMA structure, 4 steps.
// ---------------------------------------------------------------------------
__global__ __launch_bounds__(256)
void gemm2_kernel(const float* __restrict__ z, const float* __restrict__ W,
                  float* __restrict__ h, int N) {
    __shared__ float sW[FDIM * FDIM];
    if (threadIdx.x < FDIM * FDIM) sW[threadIdx.x] = W[threadIdx.x];
    __syncthreads();

    const int wave = threadIdx.x >> 5;
    const int lane = threadIdx.x & 31;
    const int ntiles = (N + 15) >> 4;
    const int tile = blockIdx.x * 8 + wave;
    if (tile >= ntiles) return;

    const int mn   = lane & 15;
    const int koff = (lane >> 4) << 1;
    int row = tile * 16 + mn; if (row > N - 1) row = N - 1;
    const float* zp = z + (size_t)row * FDIM + koff;

    v8f c = {};
    for (int k = 0; k < FDIM; k += 4) {
        v2f a = *(const v2f*)(zp + k);
        v2f b;
        b.x = sW[(k + koff)     * FDIM + mn];
        b.y = sW[(k + koff + 1) * FDIM + mn];
        c = __builtin_amdgcn_wmma_f32_16x16x4_f32(
                false, a, false, b, (short)0, c, false, false);
    }

    const int m0 = (lane < 16) ? 0 : 8;
    for (int r = 0; r < 8; ++r) {
        int gm = tile * 16 + m0 + r;
        if (gm < N) h[(size_t)gm * FDIM + mn] = c[r];
    }
}

// ---------------------------------------------------------------------------
// 5/8) edge aggregation: acc[col] += h[row] * dinv[row]*dinv[col].
//      4 threads per edge, float4 gather + 4 f32 atomics (L2-resident: h and
//      acc total ~13 MB << 192 MB L2).
// ---------------------------------------------------------------------------
__global__ void aggregate_kernel(const long long* __restrict__ ei, long long E,
                                 const float* __restrict__ dinv,
                                 const float* __restrict__ h,
                                 float* __restrict__ acc) {
    long long t = (long long)blockIdx.x * blockDim.x + threadIdx.x;
    if (t >= 4 * E) return;
    long long e = t >> 2;
    int fb  = ((int)(t & 3)) << 2;
    int row = (int)ei[e];
    int col = (int)ei[E + e];
    float norm = dinv[row] * dinv[col];
    float4 hv = *(const float4*)(h + (size_t)row * FDIM + fb);
    float* ap = acc + (size_t)col * FDIM + fb;
    atomicAdd(ap + 0, hv.x * norm);
    atomicAdd(ap + 1, hv.y * norm);
    atomicAdd(ap + 2, hv.z * norm);
    atomicAdd(ap + 3, hv.w * norm);
}

// ---------------------------------------------------------------------------
// 6) layer-1 epilogue: add self-loop term (no atomics needed), bias, ReLU.
//    In-place on the accumulator (becomes layer-2 input).
// ---------------------------------------------------------------------------
__global__ void sl_bias_relu_kernel(const float* __restrict__ h,
                                    const float* __restrict__ dinv,
                                    const float* __restrict__ b,
                                    float* __restrict__ a, int N) {
    int idx = blockIdx.x * blockDim.x + threadIdx.x;
    if (idx >= N * FDIM) return;
    int i = idx >> 4, f = idx & 15;
    float d = dinv[i];
    float v = a[idx] + h[idx] * d * d + b[f];
    a[idx] = fmaxf(v, 0.0f);
}

// ---------------------------------------------------------------------------
// 9) layer-2 epilogue: self-loop + bias, then row-wise log_softmax (16 wide).
// ---------------------------------------------------------------------------
__global__ void final_kernel(const float* __restrict__ h2,
                             const float* __restrict__ a2,
                             const float* __restrict__ dinv,
                             const float* __restrict__ b,
                             float* __restrict__ out, int N) {
    int i = blockIdx.x * blockDim.x + threadIdx.x;
    if (i >= N) return;
    float d = dinv[i];
    float dd = d * d;
    float v[FDIM];
    float m = -3.0e38f;
    size_t base = (size_t)i * FDIM;
    for (int f = 0; f < FDIM; ++f) {
        float t = a2[base + f] + h2[base + f] * dd + b[f];
        v[f] = t;
        m = fmaxf(m, t);
    }
    float s = 0.0f;
    for (int f = 0; f < FDIM; ++f) s += expf(v[f] - m);
    float ls = logf(s) + m;
    for (int f = 0; f < FDIM; ++f) out[base + f] = v[f] - ls;
}

// ---------------------------------------------------------------------------
// Launch. Inputs: x[N,512] f32, edge_index[2,E] i64, W1[512,16], b1[16],
// W2[16,16], b2[16]. Output: [N,16] f32 log-probs.
// Workspace: dinv(N) | h1(N*16) | a1(N*16) | h2(N*16) | a2(N*16)  (~26 MB)
// ---------------------------------------------------------------------------
extern "C" void kernel_launch(void* const* d_in, const int* in_sizes, int n_in,
                              void* d_out, int out_size, void* d_ws, size_t ws_size,
                              hipStream_t stream) {
    const float*     x  = (const float*)d_in[0];
    const long long* ei = (const long long*)d_in[1];
    const float*     W1 = (const float*)d_in[2];
    const float*     b1 = (const float*)d_in[3];
    const float*     W2 = (const float*)d_in[4];
    const float*     b2 = (const float*)d_in[5];
    float*           out = (float*)d_out;

    const int       N = in_sizes[0] / IN_DIM;       // 100000
    const long long E = (long long)in_sizes[1] / 2; // 3200000

    float* ws   = (float*)d_ws;
    float* dinv = ws;
    float* h1   = dinv + N;
    float* a1   = h1 + (size_t)N * FDIM;
    float* h2   = a1 + (size_t)N * FDIM;
    float* a2   = h2 + (size_t)N * FDIM;

    const int T = 256;
    const int ntiles  = (N + 15) / 16;
    const int gblocks = (ntiles + 7) / 8;

    init_ws_kernel<<<(N * FDIM + T - 1) / T, T, 0, stream>>>(dinv, a1, a2, N);
    degree_kernel<<<(int)((E + T - 1) / T), T, 0, stream>>>(ei, E, dinv);
    rsqrt_kernel<<<(N + T - 1) / T, T, 0, stream>>>(dinv, N);

    gemm1_kernel<<<gblocks, 256, 0, stream>>>(x, W1, h1, N);
    aggregate_kernel<<<(int)((4 * E + T - 1) / T), T, 0, stream>>>(ei, E, dinv, h1, a1);
    sl_bias_relu_kernel<<<(N * FDIM + T - 1) / T, T, 0, stream>>>(h1, dinv, b1, a1, N);

    gemm2_kernel<<<gblocks, 256, 0, stream>>>(a1, W2, h2, N);
    aggregate_kernel<<<(int)((4 * E + T - 1) / T), T, 0, stream>>>(ei, E, dinv, h2, a2);
    final_kernel<<<(N + T - 1) / T, T, 0, stream>>>(h2, a2, dinv, b2, out, N);
}